// LSTMRegressPose_56942676411121
// MI455X (gfx1250) — compile-verified
//
#include <hip/hip_runtime.h>
#include <stdint.h>

// Reference returns broadcast of x[:, -1, :] to (BATCH, FUTURE, N_IN).
// The GRU scans are dead code. This is a pure 16 MB streaming broadcast:
// roofline ~0.7us at 23.3 TB/s. We use the CDNA5 async data-mover path:
//   global_load_async_to_lds_b128  (stage 2KB row into LDS, once per block)
//   global_store_async_from_lds_b128 (fan out 8 rows per block, B128/lane)
// tracked with ASYNCcnt / s_wait_asynccnt.

#define N_IN       512
#define INPUT_LEN  128
#define FUTURE     32
#define BATCH      256

constexpr int THREADS           = 256;              // 8 waves (wave32)
constexpr int BLOCKS_PER_BATCH  = 4;
constexpr int ROWS_PER_BLOCK    = FUTURE / BLOCKS_PER_BATCH;   // 8
constexpr int F4_PER_ROW        = N_IN / 4;                    // 128 float4 per row
constexpr int F4_PER_BLOCK      = ROWS_PER_BLOCK * F4_PER_ROW; // 1024
constexpr int STORE_ITERS       = F4_PER_BLOCK / THREADS;      // 4

__device__ __forceinline__ uint32_t lds_offset_of(const void* p) {
    // Generic (flat) pointer to __shared__ data: addr[63:32] = SHARED_BASE,
    // addr[31:0] = workgroup-relative LDS byte offset (what async ops want).
    return (uint32_t)(uintptr_t)p;
}

__global__ __launch_bounds__(THREADS)
void gru_broadcast_last_kernel(const float* __restrict__ x,
                               float* __restrict__ out) {
    __shared__ float row[N_IN];   // 2 KB: last-timestep row for this batch

    const int tid   = threadIdx.x;
    const int blk   = blockIdx.x;
    const int b     = blk / BLOCKS_PER_BATCH;
    const int fbase = (blk % BLOCKS_PER_BATCH) * ROWS_PER_BLOCK;

    // ---- Phase 1: async DMA x[b, INPUT_LEN-1, :] (2 KB) into LDS ----------
    const float* src = x + ((size_t)b * INPUT_LEN + (INPUT_LEN - 1)) * N_IN;
    if (tid < F4_PER_ROW) {
        uint32_t lds_addr = lds_offset_of(&row[tid * 4]);
        const float* g    = src + tid * 4;
        asm volatile("global_load_async_to_lds_b128 %0, %1, off"
                     :: "v"(lds_addr), "v"(g)
                     : "memory");
    }
    asm volatile("s_wait_asynccnt 0" ::: "memory");
    __syncthreads();

    // ---- Phase 2: fan out ROWS_PER_BLOCK copies straight from LDS ---------
    float* dstbase = out + ((size_t)b * FUTURE + fbase) * N_IN;
#pragma unroll
    for (int k = 0; k < STORE_ITERS; ++k) {
        int j  = tid + k * THREADS;        // [0, F4_PER_BLOCK)
        int f  = j >> 7;                   // j / F4_PER_ROW
        int c4 = j & (F4_PER_ROW - 1);     // j % F4_PER_ROW
        uint32_t lds_addr = lds_offset_of(&row[c4 * 4]);
        float* g          = dstbase + (size_t)f * N_IN + c4 * 4;
        asm volatile("global_store_async_from_lds_b128 %0, %1, off"
                     :: "v"(g), "v"(lds_addr)
                     : "memory");
    }
    // Drain outstanding async stores before LDS is released at wave end.
    asm volatile("s_wait_asynccnt 0" ::: "memory");
}

extern "C" void kernel_launch(void* const* d_in, const int* in_sizes, int n_in,
                              void* d_out, int out_size, void* d_ws, size_t ws_size,
                              hipStream_t stream) {
    (void)in_sizes; (void)n_in; (void)d_ws; (void)ws_size; (void)out_size;
    const float* x = (const float*)d_in[0];   // (BATCH, INPUT_LEN, N_IN) f32
    float* out     = (float*)d_out;           // (BATCH, FUTURE, N_IN) f32

    const int grid = BATCH * BLOCKS_PER_BATCH;  // 1024 blocks
    gru_broadcast_last_kernel<<<grid, THREADS, 0, stream>>>(x, out);
}